// SparseConvNeXtV2_55542517072460
// MI455X (gfx1250) — compile-verified
//
#include <hip/hip_runtime.h>
#include <math.h>

typedef __attribute__((ext_vector_type(2))) float v2f;
typedef __attribute__((ext_vector_type(8))) float v8f;

// ---------------------------------------------------------------------------
// Stem: conv 4x4 stride 4 (3->32) + bias + LN(ch) ; m = 1-mask ; y *= m
// One thread per output pixel; weights (1536 f32) staged in LDS.
// ---------------------------------------------------------------------------
__global__ __launch_bounds__(256) void k_stem(
    const float* __restrict__ x, const float* __restrict__ mask,
    const float* __restrict__ sw, const float* __restrict__ sb,
    const float* __restrict__ lg, const float* __restrict__ lb,
    float* __restrict__ y, float* __restrict__ m)
{
    __shared__ float wsh[1536];
    for (int i = threadIdx.x; i < 1536; i += blockDim.x) wsh[i] = sw[i];
    __syncthreads();

    const int p  = blockIdx.x * blockDim.x + threadIdx.x;   // 0..512*512-1
    const int ph = p >> 9, pw = p & 511;

    float acc[32];
    #pragma unroll
    for (int c = 0; c < 32; ++c) acc[c] = sb[c];

    for (int kh = 0; kh < 4; ++kh)
        for (int kw = 0; kw < 4; ++kw) {
            const float* xp = x + ((size_t)(ph * 4 + kh) * 2048 + (pw * 4 + kw)) * 3;
            #pragma unroll
            for (int ci = 0; ci < 3; ++ci) {
                const float v = xp[ci];
                const float* wp = &wsh[((kh * 4 + kw) * 3 + ci) * 32];
                #pragma unroll
                for (int co = 0; co < 32; ++co) acc[co] = fmaf(v, wp[co], acc[co]);
            }
        }

    float mu = 0.f;
    #pragma unroll
    for (int c = 0; c < 32; ++c) mu += acc[c];
    mu *= (1.0f / 32.0f);
    float var = 0.f;
    #pragma unroll
    for (int c = 0; c < 32; ++c) { const float d = acc[c] - mu; var = fmaf(d, d, var); }
    var *= (1.0f / 32.0f);
    const float rs = rsqrtf(var + 1e-6f);
    const float mv = 1.0f - mask[p];
    m[p] = mv;
    float* yp = y + (size_t)p * 32;
    #pragma unroll
    for (int c = 0; c < 32; ++c)
        yp[c] = ((acc[c] - mu) * rs * lg[c] + lb[c]) * mv;
}

// ---------------------------------------------------------------------------
// Fused: depthwise 7x7 conv (SAME) + bias, *m, LN(ch), *m  ->  t
// ---------------------------------------------------------------------------
__global__ __launch_bounds__(256) void k_dw_ln(
    const float* __restrict__ y, const float* __restrict__ m,
    const float* __restrict__ dwv, const float* __restrict__ dwb,
    const float* __restrict__ lg, const float* __restrict__ lb,
    float* __restrict__ t, int H)
{
    __shared__ float wsh[1568];
    for (int i = threadIdx.x; i < 1568; i += blockDim.x) wsh[i] = dwv[i];
    __syncthreads();

    const int p  = blockIdx.x * blockDim.x + threadIdx.x;
    const int ph = p / H, pw = p - ph * H;

    float acc[32];
    #pragma unroll
    for (int c = 0; c < 32; ++c) acc[c] = dwb[c];

    for (int dh = -3; dh <= 3; ++dh) {
        const int ih = ph + dh;
        if ((unsigned)ih >= (unsigned)H) continue;
        for (int dw = -3; dw <= 3; ++dw) {
            const int iw = pw + dw;
            if ((unsigned)iw >= (unsigned)H) continue;
            const float* yp = y + ((size_t)ih * H + iw) * 32;
            const float* wp = &wsh[((dh + 3) * 7 + (dw + 3)) * 32];
            #pragma unroll
            for (int c = 0; c < 32; ++c) acc[c] = fmaf(yp[c], wp[c], acc[c]);
        }
    }

    const float mv = m[p];
    #pragma unroll
    for (int c = 0; c < 32; ++c) acc[c] *= mv;

    float mu = 0.f;
    #pragma unroll
    for (int c = 0; c < 32; ++c) mu += acc[c];
    mu *= (1.0f / 32.0f);
    float var = 0.f;
    #pragma unroll
    for (int c = 0; c < 32; ++c) { const float d = acc[c] - mu; var = fmaf(d, d, var); }
    var *= (1.0f / 32.0f);
    const float rs = rsqrtf(var + 1e-6f);
    float* tp = t + (size_t)p * 32;
    #pragma unroll
    for (int c = 0; c < 32; ++c)
        tp[c] = ((acc[c] - mu) * rs * lg[c] + lb[c]) * mv;
}

// ---------------------------------------------------------------------------
// Downsample: hld = LN(y)*m ; y' = conv 2x2 s2 (32->32) + b ; m' = maxpool2x2(m)
// One thread per output pixel; ds weights (4096 f32) in LDS.
// ---------------------------------------------------------------------------
__global__ __launch_bounds__(256) void k_down(
    const float* __restrict__ yin, const float* __restrict__ min_,
    const float* __restrict__ dsw, const float* __restrict__ dsb,
    const float* __restrict__ lg, const float* __restrict__ lb,
    float* __restrict__ yout, float* __restrict__ mout, int Ho)
{
    __shared__ float wsh[4096];
    for (int i = threadIdx.x; i < 4096; i += blockDim.x) wsh[i] = dsw[i];
    __syncthreads();

    const int p  = blockIdx.x * blockDim.x + threadIdx.x;
    const int oh = p / Ho, ow = p - oh * Ho;
    const int Hi = Ho * 2;

    float acc[32];
    #pragma unroll
    for (int c = 0; c < 32; ++c) acc[c] = dsb[c];
    float mo = 0.f;

    for (int th = 0; th < 2; ++th)
        for (int tw = 0; tw < 2; ++tw) {
            const int ih = oh * 2 + th, iw = ow * 2 + tw;
            const float* yp = yin + ((size_t)ih * Hi + iw) * 32;
            const float mi = min_[(size_t)ih * Hi + iw];
            mo = fmaxf(mo, mi);
            float vin[32];
            float mu = 0.f;
            #pragma unroll
            for (int c = 0; c < 32; ++c) { vin[c] = yp[c]; mu += vin[c]; }
            mu *= (1.0f / 32.0f);
            float var = 0.f;
            #pragma unroll
            for (int c = 0; c < 32; ++c) { const float d = vin[c] - mu; var = fmaf(d, d, var); }
            var *= (1.0f / 32.0f);
            const float rs = rsqrtf(var + 1e-6f);
            #pragma unroll
            for (int ci = 0; ci < 32; ++ci) {
                const float hv = ((vin[ci] - mu) * rs * lg[ci] + lb[ci]) * mi;
                const float* wp = &wsh[((th * 2 + tw) * 32 + ci) * 32];
                #pragma unroll
                for (int co = 0; co < 32; ++co) acc[co] = fmaf(hv, wp[co], acc[co]);
            }
        }

    float* yp = yout + (size_t)p * 32;
    #pragma unroll
    for (int c = 0; c < 32; ++c) yp[c] = acc[c] * mo;
    mout[p] = mo;
}

// ---------------------------------------------------------------------------
// MLP1 (WMMA fp32): t1[p,f] = gelu( (t[p,:] @ w1[:,f] + b1[f]) * m[p] )
// Each wave owns 8 M-tiles (128 pixels). For each of the 8 N-tiles it loads
// the B tile (w1 slice) ONCE into 8 v2f registers and sweeps all M tiles, so
// the inner loop is a pure b64-load + v_wmma stream.
// A layout (32-bit 16x4): lanes 0-15 K={k0,k0+1}, lanes 16-31 K={k0+2,k0+3}.
// ---------------------------------------------------------------------------
__global__ __launch_bounds__(256) void k_mlp1(
    const float* __restrict__ t, const float* __restrict__ w1,
    const float* __restrict__ b1, const float* __restrict__ m,
    float* __restrict__ t1)
{
    const int lane = threadIdx.x & 31;
    const int wave = blockIdx.x * 8 + (threadIdx.x >> 5);
    const int p0   = wave * 128;          // 8 M-tiles of 16 pixels
    const int row  = lane & 15;           // M row for A, N column for B/D
    const int kh   = (lane >> 4) << 1;    // 0 or 2

    for (int nt = 0; nt < 8; ++nt) {
        const int f0 = nt * 16;
        v2f b[8];
        #pragma unroll
        for (int kk = 0; kk < 8; ++kk) {
            b[kk].x = w1[(size_t)(kk * 4 + kh)     * 128 + f0 + row];
            b[kk].y = w1[(size_t)(kk * 4 + kh + 1) * 128 + f0 + row];
        }
        const float bias = b1[f0 + row];
        for (int mt = 0; mt < 8; ++mt) {
            const int pb = p0 + mt * 16;
            const float* arow = t + (size_t)(pb + row) * 32 + kh;
            v8f acc = {};
            #pragma unroll
            for (int kk = 0; kk < 8; ++kk) {
                v2f a;
                a.x = arow[kk * 4];
                a.y = arow[kk * 4 + 1];
                acc = __builtin_amdgcn_wmma_f32_16x16x4_f32(
                          false, a, false, b[kk], (short)0, acc, false, false);
            }
            #pragma unroll
            for (int r = 0; r < 8; ++r) {
                const int pr = (lane < 16) ? r : (r + 8);
                const int p  = pb + pr;
                float v = (acc[r] + bias) * m[p];
                v = 0.5f * v * (1.0f + erff(v * 0.70710678118654752f));  // exact gelu
                t1[(size_t)p * 128 + f0 + row] = v;
            }
        }
    }
}

// ---------------------------------------------------------------------------
// GRN reduce: Gx2[f] = sum_p t1[p,f]^2   (atomic partial sums)
// ---------------------------------------------------------------------------
__global__ __launch_bounds__(128) void k_grn_reduce(
    const float* __restrict__ t1, float* __restrict__ Gx2, int P)
{
    const int f = threadIdx.x;
    float s = 0.f;
    for (int p = blockIdx.x; p < P; p += gridDim.x) {
        const float v = t1[(size_t)p * 128 + f];
        s = fmaf(v, v, s);
    }
    atomicAdd(&Gx2[f], s);
}

// ---------------------------------------------------------------------------
// GRN finalize: Gx=sqrt(Gx2); Nx = Gx/(mean(Gx)+1e-6); coef[f] = g[f]*Nx+1
// ---------------------------------------------------------------------------
__global__ __launch_bounds__(128) void k_grn_final(
    const float* __restrict__ Gx2, const float* __restrict__ g,
    float* __restrict__ coef)
{
    __shared__ float red[128];
    const int f = threadIdx.x;
    const float gx = sqrtf(Gx2[f]);
    red[f] = gx;
    __syncthreads();
    for (int s = 64; s > 0; s >>= 1) {
        if (f < s) red[f] += red[f + s];
        __syncthreads();
    }
    const float mean = red[0] * (1.0f / 128.0f);
    coef[f] = g[f] * (gx / (mean + 1e-6f)) + 1.0f;
}

// ---------------------------------------------------------------------------
// Rank-1 constant of MLP2: gbw2[c] = sum_f grn_b[f] * w2[f,c]
// ---------------------------------------------------------------------------
__global__ __launch_bounds__(32) void k_gbw2(
    const float* __restrict__ gb, const float* __restrict__ w2,
    float* __restrict__ out)
{
    const int c = threadIdx.x;
    float s = 0.f;
    for (int f = 0; f < 128; ++f) s = fmaf(gb[f], w2[(size_t)f * 32 + c], s);
    out[c] = s;
}

// ---------------------------------------------------------------------------
// MLP2 (WMMA fp32), fully fused GRN + residual.
//   u = (t1*coef + gb)*m ;  y += (u @ w2 + b2)*m
// Folding: coef into B' = diag(coef)·w2 (done during B register load),
//          gb@w2 precomputed as gbw2[c]; mask is binary so m^2 = m:
//   y[p,c] += m[p] * ( t1[p,:] @ B'[:,c] + gbw2[c] + b2[c] )
// Each wave holds the 32-step B' tile in 64 VGPRs and sweeps 8 M-tiles;
// K=128 fully unrolled -> 32 back-to-back v_wmma per tile.
// ---------------------------------------------------------------------------
__global__ __launch_bounds__(256) void k_mlp2(
    const float* __restrict__ t1, const float* __restrict__ coef,
    const float* __restrict__ gbw2, const float* __restrict__ w2,
    const float* __restrict__ b2, const float* __restrict__ m,
    float* __restrict__ y)
{
    const int lane = threadIdx.x & 31;
    const int wave = blockIdx.x * 8 + (threadIdx.x >> 5);
    const int p0   = wave * 128;          // 8 M-tiles of 16 pixels
    const int row  = lane & 15;
    const int kh   = (lane >> 4) << 1;

    for (int nt = 0; nt < 2; ++nt) {
        const int c0 = nt * 16;
        v2f b[32];
        #pragma unroll
        for (int kk = 0; kk < 32; ++kk) {
            const int f = kk * 4 + kh;
            b[kk].x = coef[f]     * w2[(size_t)f       * 32 + c0 + row];
            b[kk].y = coef[f + 1] * w2[(size_t)(f + 1) * 32 + c0 + row];
        }
        const float cadd = gbw2[c0 + row] + b2[c0 + row];
        for (int mt = 0; mt < 8; ++mt) {
            const int pb = p0 + mt * 16;
            const float* arow = t1 + (size_t)(pb + row) * 128 + kh;
            v8f acc = {};
            #pragma unroll
            for (int kk = 0; kk < 32; ++kk) {
                v2f a;
                a.x = arow[kk * 4];
                a.y = arow[kk * 4 + 1];
                acc = __builtin_amdgcn_wmma_f32_16x16x4_f32(
                          false, a, false, b[kk], (short)0, acc, false, false);
            }
            #pragma unroll
            for (int r = 0; r < 8; ++r) {
                const int pr = (lane < 16) ? r : (r + 8);
                const int p  = pb + pr;
                const size_t oi = (size_t)p * 32 + c0 + row;
                y[oi] = fmaf(m[p], acc[r] + cadd, y[oi]);
            }
        }
    }
}

// ---------------------------------------------------------------------------
// Densify: NHWC (1,H,H,32) -> NCHW (1,32,H,H)
// ---------------------------------------------------------------------------
__global__ void k_transpose(const float* __restrict__ y, float* __restrict__ out, int H)
{
    const int i  = blockIdx.x * blockDim.x + threadIdx.x;
    const int hw = H * H;
    if (i >= 32 * hw) return;
    const int c = i / hw;
    const int r = i - c * hw;
    out[i] = y[(size_t)r * 32 + c];
}

// ---------------------------------------------------------------------------
extern "C" void kernel_launch(void* const* d_in, const int* in_sizes, int n_in,
                              void* d_out, int out_size, void* d_ws, size_t ws_size,
                              hipStream_t stream)
{
    (void)in_sizes; (void)n_in; (void)out_size; (void)ws_size;
    const float* x        = (const float*)d_in[0];
    const float* mask     = (const float*)d_in[1];
    const float* stem_w   = (const float*)d_in[2];
    const float* stem_b   = (const float*)d_in[3];
    const float* stem_lg  = (const float*)d_in[4];
    const float* stem_lb  = (const float*)d_in[5];
    const float* dw_w     = (const float*)d_in[6];   // (4,7,7,1,32)
    const float* dw_b     = (const float*)d_in[7];   // (4,32)
    const float* blk_lg   = (const float*)d_in[8];   // (4,32)
    const float* blk_lb   = (const float*)d_in[9];   // (4,32)
    const float* w1       = (const float*)d_in[10];  // (4,32,128)
    const float* b1       = (const float*)d_in[11];  // (4,128)
    const float* grn_g    = (const float*)d_in[12];  // (4,128)
    const float* grn_b    = (const float*)d_in[13];  // (4,128)
    const float* w2       = (const float*)d_in[14];  // (4,128,32)
    const float* b2       = (const float*)d_in[15];  // (4,32)
    const float* ds_lg    = (const float*)d_in[16];  // (3,32)
    const float* ds_lb    = (const float*)d_in[17];  // (3,32)
    const float* ds_w     = (const float*)d_in[18];  // (3,2,2,32,32)
    const float* ds_b     = (const float*)d_in[19];  // (3,32)

    // workspace layout (floats)
    float* ws   = (float*)d_ws;
    float* mA   = ws;                       // 512*512
    float* mB   = mA + 262144;              // 512*512
    float* yA   = mB + 262144;              // 512*512*32
    float* yB   = yA + 8388608;
    float* tb   = yB + 8388608;             // LN'd depthwise output (<=512*512*32)
    float* t1b  = tb + 8388608;             // hidden tensor (<=512*512*128)
    float* Gx2  = t1b + 33554432;           // 128
    float* coef = Gx2 + 128;                // 128
    float* gbw2 = coef + 128;               // 32

    // ---- stem ----
    k_stem<<<1024, 256, 0, stream>>>(x, mask, stem_w, stem_b, stem_lg, stem_lb, yA, mA);

    float* ycur = yA; float* yalt = yB;
    float* mcur = mA; float* malt = mB;
    int H = 512;

    for (int i = 0; i < 4; ++i) {
        if (i > 0) {
            const int Ho = H / 2, Po = Ho * Ho;
            k_down<<<Po / 256, 256, 0, stream>>>(
                ycur, mcur, ds_w + (size_t)(i - 1) * 4096, ds_b + (i - 1) * 32,
                ds_lg + (i - 1) * 32, ds_lb + (i - 1) * 32, yalt, malt, Ho);
            float* tswp = ycur; ycur = yalt; yalt = tswp;
            tswp = mcur; mcur = malt; malt = tswp;
            H = Ho;
        }
        const int P = H * H;
        k_dw_ln<<<P / 256, 256, 0, stream>>>(
            ycur, mcur, dw_w + (size_t)i * 1568, dw_b + i * 32,
            blk_lg + i * 32, blk_lb + i * 32, tb, H);
        k_mlp1<<<P / 1024, 256, 0, stream>>>(
            tb, w1 + (size_t)i * 4096, b1 + i * 128, mcur, t1b);
        hipMemsetAsync(Gx2, 0, 128 * sizeof(float), stream);
        k_grn_reduce<<<512, 128, 0, stream>>>(t1b, Gx2, P);
        k_grn_final<<<1, 128, 0, stream>>>(Gx2, grn_g + i * 128, coef);
        k_gbw2<<<1, 32, 0, stream>>>(grn_b + i * 128, w2 + (size_t)i * 4096, gbw2);
        k_mlp2<<<P / 1024, 256, 0, stream>>>(
            t1b, coef, gbw2, w2 + (size_t)i * 4096, b2 + i * 32,
            mcur, ycur);
    }

    // ---- densify to NCHW ----
    k_transpose<<<(32 * H * H + 255) / 256, 256, 0, stream>>>(ycur, (float*)d_out, H);
}